// DiceLoss_62208306315868
// MI455X (gfx1250) — compile-verified
//
#include <hip/hip_runtime.h>
#include <cstdint>

// Problem constants (from the reference)
#define N_IMG 16
#define C_CH  4
#define HW    768
#define M_PIX (HW * HW)          // 589824 pixels per image
#define CM    (C_CH * M_PIX)     // elements per image in predict
#define BLK_PER_IMG 72
#define PIX_PER_BLK 8192         // 72 * 8192 == 589824 exactly
#define TILE_PX 1024             // per block per tile (8 waves * 128 px)
#define TILES   8
#define WAVES   8

// ---- CDNA5 async global->LDS copy (128b per lane), GVS addressing:
//      reads from the SGPR-pair base plus the VGPR byte offset, writes to the LDS address in vdst
__device__ __forceinline__ void async_ld_b128(unsigned lds_addr,
                                              unsigned voff_bytes,
                                              const float* sbase) {
  asm volatile("global_load_async_to_lds_b128 %0, %1, %2"
               :: "v"(lds_addr), "v"(voff_bytes), "s"(sbase)
               : "memory");
}

__global__ __launch_bounds__(256) void dice_partial(
    const float* __restrict__ predict, const int* __restrict__ target,
    const int* __restrict__ masks, float* __restrict__ part)
{
  // wave-private double buffers: 8 waves * 2 bufs * (4 ch * 128 px) floats = 32 KB
  __shared__ float stage[WAVES * 2 * 512];
  __shared__ float red[WAVES][8];

  const unsigned tid  = threadIdx.x;
  const unsigned lane = tid & 31u;        // wave32
  const unsigned wv   = tid >> 5;
  const int      n    = blockIdx.y;
  const unsigned m0   = blockIdx.x * PIX_PER_BLK;

  const float* sbase = predict + (size_t)n * CM;   // uniform -> SGPR pair
  const int*   tg    = target + (size_t)n * M_PIX;
  const int*   mk    = masks  + (size_t)n * M_PIX;

  // generic pointer low 32 bits == LDS byte offset on gfx1250
  const unsigned lds_wave = (unsigned)(uintptr_t)&stage[wv * 1024];

  float anum[4] = {0.f, 0.f, 0.f, 0.f};
  float aden[4] = {0.f, 0.f, 0.f, 0.f};

  // ---- prime the pipeline: issue tile 0 into buffer 0
  {
    const unsigned mb = m0 + wv * 128u;
    #pragma unroll
    for (int c = 0; c < 4; ++c)
      async_ld_b128(lds_wave + (unsigned)((c * 128 + lane * 4) * 4),
                    (unsigned)((c * M_PIX + mb + lane * 4) * 4), sbase);
  }

  #pragma unroll
  for (int t = 0; t < TILES; ++t) {
    // issue tile t+1, then wait until tile t's 4 loads have retired
    if (t + 1 < TILES) {
      const unsigned mb = m0 + (unsigned)(t + 1) * TILE_PX + wv * 128u;
      const unsigned b  = (unsigned)(t + 1) & 1u;
      #pragma unroll
      for (int c = 0; c < 4; ++c)
        async_ld_b128(lds_wave + (b * 512u + c * 128u + lane * 4u) * 4u,
                      (unsigned)((c * M_PIX + mb + lane * 4) * 4), sbase);
      asm volatile("s_wait_asynccnt 4" ::: "memory");
    } else {
      asm volatile("s_wait_asynccnt 0" ::: "memory");
    }

    const unsigned b  = (unsigned)t & 1u;
    const unsigned mb = m0 + (unsigned)t * TILE_PX + wv * 128u;
    const float* buf  = stage + wv * 1024u + b * 512u;

    // each lane consumes the 4 pixels it staged, all 4 channels
    float pc[4][4];
    #pragma unroll
    for (int c = 0; c < 4; ++c) {
      float4 v = ((const float4*)(buf + c * 128))[lane];
      pc[c][0] = v.x; pc[c][1] = v.y; pc[c][2] = v.z; pc[c][3] = v.w;
    }
    const int4 tv = ((const int4*)(tg + mb))[lane];
    const int4 mv = ((const int4*)(mk + mb))[lane];
    const int tcl[4] = {tv.x, tv.y, tv.z, tv.w};
    const int msk[4] = {mv.x, mv.y, mv.z, mv.w};

    #pragma unroll
    for (int j = 0; j < 4; ++j) {
      const float x0 = pc[0][j], x1 = pc[1][j], x2 = pc[2][j], x3 = pc[3][j];
      const float mx = fmaxf(fmaxf(x0, x1), fmaxf(x2, x3));
      const float e0 = __expf(x0 - mx), e1 = __expf(x1 - mx);
      const float e2 = __expf(x2 - mx), e3 = __expf(x3 - mx);
      const float inv = 1.0f / (e0 + e1 + e2 + e3);
      const float sm[4] = {e0 * inv, e1 * inv, e2 * inv, e3 * inv};
      const bool on = (msk[j] >= 1);
      const int  tt = tcl[j];
      #pragma unroll
      for (int c = 0; c < 4; ++c) {
        const float oh = (c == tt) ? 1.0f : 0.0f;
        const float pe = on ? sm[c] : oh;     // p_eff
        anum[c] += pe * oh;                   // p_eff * onehot
        aden[c] += pe * pe + oh;              // p_eff^2 + onehot^2
      }
    }
  }

  // ---- wave32 shuffle reduction, then cross-wave via tiny LDS
  float vals[8] = {anum[0], anum[1], anum[2], anum[3],
                   aden[0], aden[1], aden[2], aden[3]};
  #pragma unroll
  for (int k = 0; k < 8; ++k) {
    float v = vals[k];
    #pragma unroll
    for (int off = 16; off >= 1; off >>= 1)
      v += __shfl_down(v, off, 32);
    vals[k] = v;
  }
  if (lane == 0) {
    #pragma unroll
    for (int k = 0; k < 8; ++k) red[wv][k] = vals[k];
  }
  __syncthreads();
  if (tid < 8) {
    float s = 0.f;
    #pragma unroll
    for (int w = 0; w < WAVES; ++w) s += red[w][tid];
    part[((size_t)n * BLK_PER_IMG + blockIdx.x) * 8 + tid] = s;
  }
}

// Fold 1152 block-partials into the scalar loss, deterministically.
__global__ void dice_finalize(const float* __restrict__ part,
                              float* __restrict__ out)
{
  __shared__ float r[64];
  const int t = threadIdx.x;  // blockDim.x == 64
  float loss = 0.f;
  {
    const int n = t >> 2, c = t & 3;
    float num = 1.0f, den = 1.0f;   // SMOOTH = 1
    for (int b = 0; b < BLK_PER_IMG; ++b) {
      const float* p = part + ((size_t)(n * BLK_PER_IMG + b)) * 8;
      num += p[c];
      den += p[4 + c];
    }
    loss = 1.0f - num / den;
  }
  r[t] = loss;
  __syncthreads();
  if (t == 0) {
    float s = 0.f;
    for (int i = 0; i < 64; ++i) s += r[i];
    out[0] = s / (float)(N_IMG * C_CH);   // mean over batch, /C over classes
  }
}

extern "C" void kernel_launch(void* const* d_in, const int* in_sizes, int n_in,
                              void* d_out, int out_size, void* d_ws, size_t ws_size,
                              hipStream_t stream) {
  (void)in_sizes; (void)n_in; (void)out_size; (void)ws_size;
  const float* predict = (const float*)d_in[0];
  const int*   target  = (const int*)d_in[1];
  const int*   masks   = (const int*)d_in[2];
  float*       part    = (float*)d_ws;     // 16*72*8 floats = 36,864 B
  float*       out     = (float*)d_out;

  dim3 grid(BLK_PER_IMG, N_IMG);
  dice_partial<<<grid, 256, 0, stream>>>(predict, target, masks, part);
  dice_finalize<<<1, 64, 0, stream>>>(part, out);
}